// HydraGNN_7773890806311
// MI455X (gfx1250) — compile-verified
//
#include <hip/hip_runtime.h>

typedef __attribute__((ext_vector_type(2))) float v2f;
typedef __attribute__((ext_vector_type(8))) float v8f;

// ---------------------------------------------------------------------------
// Zero-fill scratch, float4-vectorized (agg1, agg2, deg contiguous in ws)
// ---------------------------------------------------------------------------
__global__ void zero_f32(float* __restrict__ p, long n) {
    long i4 = (long)blockIdx.x * blockDim.x + threadIdx.x;
    long stride = (long)gridDim.x * blockDim.x;
    long n4 = n >> 2;
    float4* p4 = (float4*)p;
    const float4 z = make_float4(0.f, 0.f, 0.f, 0.f);
    for (long i = i4; i < n4; i += stride) p4[i] = z;
    // tail
    for (long i = (n4 << 2) + i4; i < n; i += stride) p[i] = 0.0f;
}

// ---------------------------------------------------------------------------
// Edge aggregation: agg[dst] += feat[src] (64 channels), optional degree.
// One wave (32 lanes) per edge; lane L handles channels L and L+32.
// feat / agg are L2-resident (12.8 MB each) so atomics resolve in L2.
// ---------------------------------------------------------------------------
__global__ void edge_agg64(const float* __restrict__ feat,
                           const int* __restrict__ src,
                           const int* __restrict__ dst,
                           float* __restrict__ agg,
                           float* __restrict__ deg,
                           int E) {
    const int lane = threadIdx.x & 31;
    const int e = blockIdx.x * (blockDim.x >> 5) + (threadIdx.x >> 5);
    if (e >= E) return;
    const int s = src[e];
    const int d = dst[e];
    const float* fs = feat + (size_t)s * 64;
    float* ad = agg + (size_t)d * 64;
    __hip_atomic_fetch_add(&ad[lane],      fs[lane],      __ATOMIC_RELAXED, __HIP_MEMORY_SCOPE_AGENT);
    __hip_atomic_fetch_add(&ad[lane + 32], fs[lane + 32], __ATOMIC_RELAXED, __HIP_MEMORY_SCOPE_AGENT);
    if (lane == 0 && deg != nullptr)
        __hip_atomic_fetch_add(&deg[d], 1.0f, __ATOMIC_RELAXED, __HIP_MEMORY_SCOPE_AGENT);
}

// ---------------------------------------------------------------------------
// Fused SAGEConv GEMM via V_WMMA_F32_16X16X4_F32 (wave32, fp32 end to end):
//   out[16xM tile] = relu( (agg/deg) @ wl + xroot @ wr + bias )
// One wave computes one 16x16 output tile; block covers one 16-row stripe
// with (M/16) waves. K and M are compile-time so all fragment loads are
// base + immediate-offset; fragments are batch-loaded per term, then the
// 16 WMMAs issue back-to-back.
//
// Fragment layouts (ISA 7.12.2, wave32):
//   A 16x4 f32 : lane l -> row (l&15), K = (l>>4)*2 + {0,1}
//   B  4x16 f32: lane l -> col (l&15), K = (l>>4)*2 + {0,1}
//   C/D 16x16  : lane l -> col (l&15), rows (l>>4)*8 + v, v=0..7
// ---------------------------------------------------------------------------
template <int K, int M>
__global__ void sage_gemm_wmma(const float* __restrict__ agg,
                               const float* __restrict__ deg,
                               const float* __restrict__ xroot,
                               const float* __restrict__ wl,
                               const float* __restrict__ wr,
                               const float* __restrict__ bias,
                               float* __restrict__ out,
                               int N) {
    constexpr int S = K / 4;          // 16 K-steps
    const int lane = threadIdx.x & 31;
    const int wave = threadIdx.x >> 5;
    const int lo   = lane & 15;
    const int half = lane >> 4;
    const int row0 = blockIdx.x * 16;
    const int col0 = wave * 16;

    // A-fragment row for this lane (clamped so EXEC stays all-ones).
    int ar = row0 + lo;
    if (ar >= N) ar = N - 1;
    const float rinv = 1.0f / fmaxf(deg[ar], 1.0f);
    const int bcol = col0 + lo;       // B-fragment / bias / store column

    // Single base pointer per stream; all step offsets are immediates.
    const float* __restrict__ pa = agg   + (size_t)ar * K + half * 2;
    const float* __restrict__ px = xroot + (size_t)ar * K + half * 2;
    const float* __restrict__ pl = wl + (size_t)(half * 2) * M + bcol;
    const float* __restrict__ pr = wr + (size_t)(half * 2) * M + bcol;

    v2f af[S], bf[S];
    v8f acc = {};

    // ---- Term 1: mean-aggregated neighbors @ wl ----
#pragma unroll
    for (int s = 0; s < S; ++s) {
        af[s].x = pa[s * 4];
        af[s].y = pa[s * 4 + 1];
        bf[s].x = pl[(size_t)(s * 4) * M];
        bf[s].y = pl[(size_t)(s * 4 + 1) * M];
    }
#pragma unroll
    for (int s = 0; s < S; ++s) {
        v2f a;
        a.x = af[s].x * rinv;
        a.y = af[s].y * rinv;
        acc = __builtin_amdgcn_wmma_f32_16x16x4_f32(false, a, false, bf[s],
                                                    (short)0, acc, false, false);
    }

    // ---- Term 2: root features @ wr ----
#pragma unroll
    for (int s = 0; s < S; ++s) {
        af[s].x = px[s * 4];
        af[s].y = px[s * 4 + 1];
        bf[s].x = pr[(size_t)(s * 4) * M];
        bf[s].y = pr[(size_t)(s * 4 + 1) * M];
    }
#pragma unroll
    for (int s = 0; s < S; ++s) {
        acc = __builtin_amdgcn_wmma_f32_16x16x4_f32(false, af[s], false, bf[s],
                                                    (short)0, acc, false, false);
    }

    const float bv = bias[bcol];
#pragma unroll
    for (int v = 0; v < 8; ++v) {
        const int r = row0 + half * 8 + v;
        if (r < N) {
            float val = acc[v] + bv;
            out[(size_t)r * M + bcol] = fmaxf(val, 0.0f);
        }
    }
}

// ---------------------------------------------------------------------------
// Classifier head: h2[N,32] -> relu(h2@wc1+bc1)[N,16] -> (@wc2+bc2)[N,2].
// ~550 FLOP per node: scalar VALU is ample; weights broadcast from L2.
// ---------------------------------------------------------------------------
__global__ void classifier_head(const float* __restrict__ h2,
                                const float* __restrict__ wc1,
                                const float* __restrict__ bc1,
                                const float* __restrict__ wc2,
                                const float* __restrict__ bc2,
                                float* __restrict__ out, int N) {
    const int n = blockIdx.x * blockDim.x + threadIdx.x;
    if (n >= N) return;
    const float* h = h2 + (size_t)n * 32;
    float hn[32];
#pragma unroll
    for (int k = 0; k < 32; ++k) hn[k] = h[k];
    float h3[16];
#pragma unroll
    for (int j = 0; j < 16; ++j) {
        float s = bc1[j];
#pragma unroll
        for (int k = 0; k < 32; ++k) s = fmaf(hn[k], wc1[k * 16 + j], s);
        h3[j] = fmaxf(s, 0.0f);
    }
#pragma unroll
    for (int j = 0; j < 2; ++j) {
        float s = bc2[j];
#pragma unroll
        for (int k = 0; k < 16; ++k) s = fmaf(h3[k], wc2[k * 2 + j], s);
        out[(size_t)n * 2 + j] = s;
    }
}

// ---------------------------------------------------------------------------
// Host-side orchestration (graph-capture safe: kernels only, all on stream)
// ---------------------------------------------------------------------------
extern "C" void kernel_launch(void* const* d_in, const int* in_sizes, int n_in,
                              void* d_out, int out_size, void* d_ws, size_t ws_size,
                              hipStream_t stream) {
    const float* x    = (const float*)d_in[0];
    const int*   ei   = (const int*)d_in[1];
    const float* w1l  = (const float*)d_in[2];
    const float* w1r  = (const float*)d_in[3];
    const float* b1   = (const float*)d_in[4];
    const float* w2l  = (const float*)d_in[5];
    const float* w2r  = (const float*)d_in[6];
    const float* b2   = (const float*)d_in[7];
    const float* wc1  = (const float*)d_in[8];
    const float* bc1  = (const float*)d_in[9];
    const float* wc2  = (const float*)d_in[10];
    const float* bc2  = (const float*)d_in[11];
    float* out = (float*)d_out;

    const int N = in_sizes[0] / 64;   // 50000
    const int E = in_sizes[1] / 2;    // 800000
    const int* src = ei;
    const int* dst = ei + E;

    // Workspace layout (floats): agg1[N*64] agg2[N*64] deg[N] | h1[N*64] h2[N*32]
    float* agg1 = (float*)d_ws;
    float* agg2 = agg1 + (size_t)N * 64;
    float* deg  = agg2 + (size_t)N * 64;
    float* h1   = deg  + (size_t)N;
    float* h2   = h1   + (size_t)N * 64;

    // 1) Zero the accumulated region (agg1 + agg2 + deg, contiguous).
    {
        long nz = (long)N * (64 + 64 + 1);
        zero_f32<<<2048, 256, 0, stream>>>(agg1, nz);
    }

    // 2) Layer-1 edge aggregation + degree. 8 edges/block (one wave each).
    {
        int blocks = (E + 7) / 8;
        edge_agg64<<<blocks, 256, 0, stream>>>(x, src, dst, agg1, deg, E);
    }

    // 3) SAGE layer 1: [N,64] -> [N,64], 4 col-tiles per 16-row stripe.
    {
        int blocks = (N + 15) / 16;
        sage_gemm_wmma<64, 64><<<blocks, 128, 0, stream>>>(agg1, deg, x, w1l,
                                                           w1r, b1, h1, N);
    }

    // 4) Layer-2 edge aggregation on h1 (degree already computed).
    {
        int blocks = (E + 7) / 8;
        edge_agg64<<<blocks, 256, 0, stream>>>(h1, src, dst, agg2, nullptr, E);
    }

    // 5) SAGE layer 2: [N,64] -> [N,32], 2 col-tiles per stripe.
    {
        int blocks = (N + 15) / 16;
        sage_gemm_wmma<64, 32><<<blocks, 64, 0, stream>>>(agg2, deg, h1, w2l,
                                                          w2r, b2, h2, N);
    }

    // 6) Classifier head -> d_out [N,2].
    {
        int blocks = (N + 255) / 256;
        classifier_head<<<blocks, 256, 0, stream>>>(h2, wc1, bc1, wc2, bc2, out, N);
    }
}